// LensCrack_42906723287186
// MI455X (gfx1250) — compile-verified
//
#include <hip/hip_runtime.h>
#include <cstdint>

// LensCrack: out = clip(where(bresenham_mask, 0.05, x), 0, 1)
// x: [16,3,1024,1024] f32, endpoints: [16,8,4] i32 rows (y0,x0,y1,x1).
//
// Bandwidth-bound: 403 MB total HBM traffic -> ~17.3 us floor @ 23.3 TB/s.
// Kernel 1: streaming clamp using gfx1250 async global->LDS b128 copies,
//   4 deep per wave over 8 round-robin LDS slots (ASYNCcnt pipelined,
//   barrier-free: each lane consumes its own slot), NT b128 stores.
// Kernel 2: tiny raster kernel overwrites crack pixels with 0.05.

typedef float f32x4 __attribute__((ext_vector_type(4)));

#define TPB   256
#define NSLOT 8   // LDS slots per thread (power of two)
#define DEPTH 4   // async loads kept in flight per wave

#define CRACK_VAL 0.05f

__global__ __launch_bounds__(TPB) void lenscrack_clamp_stream(
    const f32x4* __restrict__ in, f32x4* __restrict__ out, long long n4)
{
    // 8 slots x 256 threads x 16 B = 32 KB LDS.
    __shared__ f32x4 buf[NSLOT * TPB];
    const int tid = threadIdx.x;

    // Raw 32-bit LDS byte address of this thread's slot 0 (generic pointer
    // to LDS carries the LDS offset in its low 32 bits).
    const unsigned ldsBase   = (unsigned)(uintptr_t)(&buf[tid]);
    const unsigned slotBytes = TPB * (unsigned)sizeof(f32x4);  // 4096

    long long i = (long long)blockIdx.x * TPB + tid;
    const long long stride = (long long)gridDim.x * TPB;
    if (i >= n4) return;

    // Prime the pipeline: up to DEPTH async copies in flight.
    long long ip = i;
#pragma unroll
    for (unsigned k = 0; k < DEPTH; ++k) {
        if (ip < n4) {
            unsigned long long g = (unsigned long long)(uintptr_t)(in + ip);
            unsigned lds = ldsBase + k * slotBytes;
            asm volatile("global_load_async_to_lds_b128 %0, %1, off"
                         :: "v"(lds), "v"(g) : "memory");
            ip += stride;
        }
    }

    unsigned k = 0;
    for (; i < n4; i += stride, ++k) {
        if (ip < n4) {
            // Refill slot (k+DEPTH) mod NSLOT; that slot was consumed
            // DEPTH iterations ago, so its ds_load has long completed.
            unsigned long long g = (unsigned long long)(uintptr_t)(in + ip);
            unsigned lds = ldsBase + ((k + DEPTH) & (NSLOT - 1)) * slotBytes;
            asm volatile("global_load_async_to_lds_b128 %0, %1, off"
                         :: "v"(lds), "v"(g) : "memory");
            ip += stride;
            // 5 outstanding after issue; <=4 left => oldest (slot k) landed.
            asm volatile("s_wait_asynccnt 4" ::: "memory");
        } else {
            // Drain phase: wait for everything still in flight.
            asm volatile("s_wait_asynccnt 0" ::: "memory");
        }

        // Consume own slot (issuing wave == consuming wave: ASYNCcnt
        // ordering suffices, no barrier). ds_load_b128.
        f32x4 v = buf[(k & (NSLOT - 1)) * TPB + tid];

        f32x4 r;
        r.x = fminf(fmaxf(v.x, 0.0f), 1.0f);
        r.y = fminf(fmaxf(v.y, 0.0f), 1.0f);
        r.z = fminf(fmaxf(v.z, 0.0f), 1.0f);
        r.w = fminf(fmaxf(v.w, 0.0f), 1.0f);

        // Output is never re-read: non-temporal b128 store.
        __builtin_nontemporal_store(r, out + i);
    }
}

// One thread per crack line. Replays the reference's Bresenham scan exactly:
// record (cx,cy) at step t (t <= nsteps), then update err/cx/cy using the
// same e2 conditions. Overwrites 0.05 into all 3 channels of that pixel.
__global__ void lenscrack_raster(const int* __restrict__ ep,
                                 float* __restrict__ out,
                                 int B, int K, int C, int H, int W)
{
    const int idx = blockIdx.x * blockDim.x + threadIdx.x;
    if (idx >= B * K) return;
    const int b = idx / K;

    int y0 = ep[idx * 4 + 0];
    int x0 = ep[idx * 4 + 1];
    const int y1 = ep[idx * 4 + 2];
    const int x1 = ep[idx * 4 + 3];

    const int dx = abs(x1 - x0);
    const int dy = abs(y1 - y0);
    const int sx = (x0 < x1) ? 1 : -1;
    const int sy = (y0 < y1) ? 1 : -1;
    int err = dx - dy;
    const int nsteps = max(dx, dy);

    const long long plane = (long long)H * W;
    const long long base = (long long)b * C * plane;

    for (int t = 0; t <= nsteps; ++t) {
        if (x0 >= 0 && x0 < W && y0 >= 0 && y0 < H) {
            const long long p = base + (long long)y0 * W + x0;
            out[p]             = CRACK_VAL;
            out[p + plane]     = CRACK_VAL;
            out[p + 2 * plane] = CRACK_VAL;
        }
        const int e2 = 2 * err;
        const bool cx = e2 > -dy;
        const bool cy = e2 < dx;
        err += (cy ? dx : 0) - (cx ? dy : 0);
        x0 += cx ? sx : 0;
        y0 += cy ? sy : 0;
    }
}

extern "C" void kernel_launch(void* const* d_in, const int* in_sizes, int n_in,
                              void* d_out, int out_size, void* d_ws, size_t ws_size,
                              hipStream_t stream) {
    const float* x   = (const float*)d_in[0];
    const int*   ep  = (const int*)d_in[1];
    float*       out = (float*)d_out;

    // Shapes per reference setup_inputs().
    const int B = 16, C = 3, H = 1024, W = 1024;
    const int K = in_sizes[1] / (B * 4);   // = 8

    const long long n  = (long long)out_size;  // B*C*H*W (divisible by 4)
    const long long n4 = n / 4;

    // 4096 blocks -> 12 grid-stride iterations per thread (n4/TPB = 49152),
    // amortizing the 4-deep pipeline prime; block-uniform trip counts since
    // n4 % TPB == 0.
    long long blocksNeeded = (n4 + TPB - 1) / TPB;
    int grid = (blocksNeeded < 4096) ? (int)blocksNeeded : 4096;
    lenscrack_clamp_stream<<<grid, TPB, 0, stream>>>(
        (const f32x4*)x, (f32x4*)out, n4);

    const int lines = B * K;  // 128
    lenscrack_raster<<<(lines + 127) / 128, 128, 0, stream>>>(
        ep, out, B, K, C, H, W);
}